// GraphSAGEModel_34600256537253
// MI455X (gfx1250) — compile-verified
//
#include <hip/hip_runtime.h>

typedef __attribute__((ext_vector_type(2))) float v2f;
typedef __attribute__((ext_vector_type(8))) float v8f;

#define FDIM 128   // feature dim for both layers (IN == H == 128)
#define WPB 4      // waves per block in the WMMA GEMM kernels

// ---------------------------------------------------------------------------
// Degree count: deg[dst] += 1 per edge
// ---------------------------------------------------------------------------
__global__ void deg_kernel(const long long* __restrict__ ei,
                           float* __restrict__ deg, int E) {
    int e = blockIdx.x * blockDim.x + threadIdx.x;
    if (e < E) {
        int d = (int)ei[(size_t)E + e];
        atomicAdd(&deg[d], 1.0f);
    }
}

// deg -> 1/max(deg,1), in place
__global__ void invdeg_kernel(float* __restrict__ deg, int N) {
    int i = blockIdx.x * blockDim.x + threadIdx.x;
    if (i < N) deg[i] = 1.0f / fmaxf(deg[i], 1.0f);
}

// ---------------------------------------------------------------------------
// Edge scatter: agg[dst] += feat[src].  One edge per 32 threads, 4 f32/lane.
// ---------------------------------------------------------------------------
__global__ void scatter_kernel(const float* __restrict__ feat,
                               const long long* __restrict__ ei,
                               float* __restrict__ agg, int E) {
    int t = blockIdx.x * blockDim.x + threadIdx.x;
    if (t >= E * 32) return;
    int e = t >> 5;
    int c = (t & 31) * 4;
    int s = (int)ei[e];
    int d = (int)ei[(size_t)E + e];
    const float4 v = *(const float4*)(feat + (size_t)s * FDIM + c);
    float* dp = agg + (size_t)d * FDIM + c;
    atomicAdd(dp + 0, v.x);
    atomicAdd(dp + 1, v.y);
    atomicAdd(dp + 2, v.z);
    atomicAdd(dp + 3, v.w);
}

// ---------------------------------------------------------------------------
// Fused SAGE layer GEMM (fp32 WMMA, K = 128):
//   out = act( scale(agg, invdeg) @ Wl  +  xin @ Wr  + bias )
// One wave computes a 16-row x (NT*16)-col output tile.
// NT = number of 16-wide column tiles (8 -> 128 cols, 4 -> 64 cols).
// ---------------------------------------------------------------------------
template <int NT, bool HAS_AGG, bool RELU>
__global__ __launch_bounds__(WPB * 32) void sage_gemm(
    const float* __restrict__ agg, const float* __restrict__ xin,
    const float* __restrict__ Wl,  const float* __restrict__ Wr,
    const float* __restrict__ bias, const float* __restrict__ invdeg,
    float* __restrict__ out) {
    __shared__ float lds[WPB][16 * FDIM];

    const int lane  = threadIdx.x & 31;
    const int wave  = threadIdx.x >> 5;
    const int tile  = blockIdx.x * WPB + wave;
    const int node0 = tile * 16;
    float* tl = lds[wave];

    const int row16 = lane & 15;  // A: M index / B: N index / D: column
    const int khalf = lane >> 4;  // which K pair of the 4-deep WMMA step
    const int ON    = NT * 16;    // output row stride

    v8f acc[NT];
#pragma unroll
    for (int j = 0; j < NT; ++j) acc[j] = {0, 0, 0, 0, 0, 0, 0, 0};

#pragma unroll
    for (int pass = 0; pass < 2; ++pass) {
        if (pass == 0 && !HAS_AGG) continue;
        const float* A = (pass == 0) ? agg : xin;
        const float* W = (pass == 0) ? Wl : Wr;

        // Stage this wave's 16x128 A tile into LDS (coalesced float4),
        // applying 1/deg scaling for the aggregate pass.
#pragma unroll
        for (int it = 0; it < 16; ++it) {
            int vi = it * 32 + lane;   // float4 index within tile
            int r  = vi >> 5;          // 32 float4 per 128-float row
            int c  = (vi & 31) * 4;
            float4 v = *(const float4*)(A + (size_t)(node0 + r) * FDIM + c);
            if (HAS_AGG && pass == 0) {
                float s = invdeg[node0 + r];
                v.x *= s; v.y *= s; v.z *= s; v.w *= s;
            }
            *(float4*)(tl + r * FDIM + c) = v;
        }
        __syncthreads();

        for (int k0 = 0; k0 < FDIM; k0 += 4) {
            // A fragment 16x4: lane(0-15)=row M; reg pair = K = k0+2*khalf+{0,1}
            v2f a = *(const v2f*)(tl + row16 * FDIM + k0 + 2 * khalf);
#pragma unroll
            for (int j = 0; j < NT; ++j) {
                // B fragment 4x16: lane(0-15)=col N; reg pair = same K pair
                v2f b;
                b.x = W[(size_t)(k0 + 2 * khalf)     * ON + j * 16 + row16];
                b.y = W[(size_t)(k0 + 2 * khalf + 1) * ON + j * 16 + row16];
                acc[j] = __builtin_amdgcn_wmma_f32_16x16x4_f32(
                    false, a, false, b, (short)0, acc[j], false, false);
            }
        }
        __syncthreads();
    }

    // D layout: VGPR r -> row (r + 8*khalf), col = j*16 + row16
#pragma unroll
    for (int j = 0; j < NT; ++j) {
#pragma unroll
        for (int r = 0; r < 8; ++r) {
            int m   = r + 8 * khalf;
            int col = j * 16 + row16;
            float v = acc[j][r] + bias[col];
            if (RELU) v = fmaxf(v, 0.0f);
            out[(size_t)(node0 + m) * ON + col] = v;
        }
    }
}

// ---------------------------------------------------------------------------
// Launch
// ---------------------------------------------------------------------------
static inline int cdiv(int a, int b) { return (a + b - 1) / b; }

extern "C" void kernel_launch(void* const* d_in, const int* in_sizes, int n_in,
                              void* d_out, int out_size, void* d_ws, size_t ws_size,
                              hipStream_t stream) {
    const float*     x    = (const float*)d_in[0];
    const long long* ei   = (const long long*)d_in[1];   // int64 [2, E]
    const float*     W1l  = (const float*)d_in[2];
    const float*     W1r  = (const float*)d_in[3];
    const float*     b1   = (const float*)d_in[4];
    const float*     W2l  = (const float*)d_in[5];
    const float*     W2r  = (const float*)d_in[6];
    const float*     b2   = (const float*)d_in[7];
    const float*     Wout = (const float*)d_in[8];
    const float*     bout = (const float*)d_in[9];
    float*           out  = (float*)d_out;

    const int N = in_sizes[0] / FDIM;   // 40000
    const int E = in_sizes[1] / 2;      // 640000

    // Workspace: deg | AGG | H1 | H2   (~62 MB)
    char*  ws        = (char*)d_ws;
    size_t degBytes  = (((size_t)N * sizeof(float)) + 255) & ~(size_t)255;
    size_t featBytes = ((size_t)N * FDIM * sizeof(float) + 255) & ~(size_t)255;
    float* deg = (float*)ws;
    float* AGG = (float*)(ws + degBytes);
    float* H1  = (float*)(ws + degBytes + featBytes);
    float* H2  = (float*)(ws + degBytes + 2 * featBytes);

    const int gemmBlocks = N / (16 * WPB);   // 40000/64 = 625, exact

    // Degrees (shared by both layers)
    hipMemsetAsync(deg, 0, (size_t)N * sizeof(float), stream);
    deg_kernel<<<cdiv(E, 256), 256, 0, stream>>>(ei, deg, E);
    invdeg_kernel<<<cdiv(N, 256), 256, 0, stream>>>(deg, N);

    // Layer 1
    hipMemsetAsync(AGG, 0, (size_t)N * FDIM * sizeof(float), stream);
    scatter_kernel<<<cdiv(E * 32, 256), 256, 0, stream>>>(x, ei, AGG, E);
    sage_gemm<8, true, true><<<gemmBlocks, WPB * 32, 0, stream>>>(
        AGG, x, W1l, W1r, b1, deg, H1);

    // Layer 2
    hipMemsetAsync(AGG, 0, (size_t)N * FDIM * sizeof(float), stream);
    scatter_kernel<<<cdiv(E * 32, 256), 256, 0, stream>>>(H1, ei, AGG, E);
    sage_gemm<8, true, true><<<gemmBlocks, WPB * 32, 0, stream>>>(
        AGG, H1, W2l, W2r, b2, deg, H2);

    // Output projection: out = H2 @ Wout + bout  (64 cols, no relu)
    sage_gemm<4, false, false><<<gemmBlocks, WPB * 32, 0, stream>>>(
        nullptr, H2, nullptr, Wout, bout, nullptr, out);
}